// COTREC_13331578487507
// MI455X (gfx1250) — compile-verified
//
#include <hip/hip_runtime.h>
#include <math.h>

// ---- problem constants (match reference) ----
#define N_NODE 200000
#define EMB    100
#define PE     112          // EMB padded to multiple of 16 for WMMA tiles
#define NNZ_   1000000
#define BATCH  1024
#define SEQL   50
#define TAU    0.2f

typedef float v2f __attribute__((ext_vector_type(2)));
typedef float v8f __attribute__((ext_vector_type(8)));

// ---------------- utility kernels ----------------
__global__ void k_memset(float* p, size_t n) {
  size_t i = (size_t)blockIdx.x * blockDim.x + threadIdx.x;
  size_t st = (size_t)gridDim.x * blockDim.x;
  for (; i < n; i += st) p[i] = 0.f;
}

// hyper = (emb + h0 + h1) / 3
__global__ void k_combine(const float* __restrict__ e, const float* __restrict__ h0,
                          const float* __restrict__ h1, float* __restrict__ out, size_t n) {
  size_t i = (size_t)blockIdx.x * blockDim.x + threadIdx.x;
  size_t st = (size_t)gridDim.x * blockDim.x;
  for (; i < n; i += st) out[i] = (e[i] + h0[i] + h1[i]) * (1.f / 3.f);
}

// dst[dr,dc] = src zero-padded
__global__ void k_pad_copy(const float* __restrict__ src, float* __restrict__ dst,
                           int sr, int sc, int dr, int dc) {
  int idx = blockIdx.x * blockDim.x + threadIdx.x;
  if (idx >= dr * dc) return;
  int r = idx / dc, c = idx % dc;
  dst[idx] = (r < sr && c < sc) ? src[r * sc + c] : 0.f;
}

// ---------------- HyperConv SpMM (HBM-bound, atomic scatter) ----------------
__global__ void k_spmm(const float* __restrict__ val, const int* __restrict__ row,
                       const int* __restrict__ col, const float* __restrict__ hin,
                       float* __restrict__ hout) {
  int e = blockIdx.x;         // one nonzero per block
  int t = threadIdx.x;        // 128 threads cover EMB=100
  if (t >= EMB) return;
  float v = val[e];
  int c = col[e], r = row[e];
  atomicAdd(&hout[(size_t)r * EMB + t], v * hin[(size_t)c * EMB + t]);
}

// ---------------- RelationGAT: gather + mean over session ----------------
__global__ void k_gather_mean(const float* __restrict__ hyper, const int* __restrict__ items,
                              const float* __restrict__ slen, float* __restrict__ rel) {
  int b = blockIdx.x, e = threadIdx.x;
  if (e >= PE) return;
  float s = 0.f;
  if (e < EMB) {
    for (int l = 0; l < SEQL; ++l) {
      int it = items[b * SEQL + l];
      if (it > 0) s += hyper[(size_t)(it - 1) * EMB + e];   // emb_pad row 0 = zeros
    }
    s /= slen[b];
  }
  rel[b * PE + e] = s;                                      // zero pad cols
}

// ---------------- fp32 WMMA GEMM: C[M,N] = A[M,K] @ B[K,N] (row major) ----------------
// one wave32 per 16x16 C tile; V_WMMA_F32_16X16X4_F32, K stepped by 4.
__global__ void k_gemm_nn(const float* __restrict__ A, const float* __restrict__ B,
                          float* __restrict__ C, int M, int N, int K,
                          int lda, int ldb, int ldc) {
  int lane = threadIdx.x;                 // 0..31
  int half = lane >> 4, l16 = lane & 15;
  int m0 = blockIdx.x * 16, n0 = blockIdx.y * 16;
  (void)M; (void)N;
  v8f acc = {};
  const float* arow = A + (size_t)(m0 + l16) * lda;
  for (int kb = 0; kb < K; kb += 4) {
    v2f a, b;
    // A 16x4 layout: VGPR i <- K = kb + 2*half + i, row = m0+l16
    a[0] = arow[kb + 2 * half + 0];
    a[1] = arow[kb + 2 * half + 1];
    // B 4x16 layout: VGPR i <- row kb + 2*half + i, col = n0+l16
    b[0] = B[(size_t)(kb + 2 * half + 0) * ldb + n0 + l16];
    b[1] = B[(size_t)(kb + 2 * half + 1) * ldb + n0 + l16];
    acc = __builtin_amdgcn_wmma_f32_16x16x4_f32(false, a, false, b, (short)0, acc,
                                                false, false);
  }
  // C/D layout: VGPR r -> row r + 8*half, col l16
  for (int r = 0; r < 8; ++r)
    C[(size_t)(m0 + r + 8 * half) * ldc + n0 + l16] = acc[r];
}

// C[M,N] = A[M,K] @ Bt[N,K]^T  (Bt row-major [N,K]) — contiguous B-pair loads per lane
__global__ void k_gemm_nt(const float* __restrict__ A, const float* __restrict__ Bt,
                          float* __restrict__ C, int M, int N, int K,
                          int lda, int ldbt, int ldc) {
  int lane = threadIdx.x;
  int half = lane >> 4, l16 = lane & 15;
  int m0 = blockIdx.x * 16, n0 = blockIdx.y * 16;
  (void)M; (void)N;
  v8f acc = {};
  const float* arow = A + (size_t)(m0 + l16) * lda;
  const float* brow = Bt + (size_t)(n0 + l16) * ldbt;
  for (int kb = 0; kb < K; kb += 4) {
    v2f a, b;
    a[0] = arow[kb + 2 * half + 0];
    a[1] = arow[kb + 2 * half + 1];
    b[0] = brow[kb + 2 * half + 0];
    b[1] = brow[kb + 2 * half + 1];
    acc = __builtin_amdgcn_wmma_f32_16x16x4_f32(false, a, false, b, (short)0, acc,
                                                false, false);
  }
  for (int r = 0; r < 8; ++r)
    C[(size_t)(m0 + r + 8 * half) * ldc + n0 + l16] = acc[r];
}

// ---------------- selu + L2 normalize -> l_c (compact) + sess (padded) + norms ----------------
__global__ void k_selu_norm(const float* __restrict__ rel2, float* __restrict__ lc,
                            float* __restrict__ sess, float* __restrict__ nrm) {
  __shared__ float sh[128];
  __shared__ float srow[PE];
  int b = blockIdx.x, t = threadIdx.x;    // 128 threads
  const float SA = 1.6732632423543772f, SL = 1.0507009873554805f;
  float x = 0.f;
  if (t < PE) {
    float v = rel2[b * PE + t];
    x = SL * (v > 0.f ? v : SA * (__expf(v) - 1.f));
    srow[t] = x;
  }
  sh[t] = (t < EMB) ? x * x : 0.f;
  __syncthreads();
  for (int s = 64; s > 0; s >>= 1) { if (t < s) sh[t] += sh[t + s]; __syncthreads(); }
  float ss = sh[0];
  float inv = 1.f / sqrtf(ss);
  if (t < PE) {
    float v = (t < EMB) ? srow[t] * inv : 0.f;
    sess[b * PE + t] = v;
    if (t < EMB) lc[b * EMB + t] = v;
  }
  if (t == 0) nrm[b] = sqrtf(ss * inv * inv + (float)EMB * 1e-6f); // sqrt(sum(s^2+1e-6))
}

// scale cos by 1/(n_i n_j) then row softmax, in place on G[B,B]
__global__ void k_cos_softmax(float* __restrict__ G, const float* __restrict__ nrm) {
  __shared__ float sh[256];
  int i = blockIdx.x, t = threadIdx.x;
  float ni = nrm[i];
  float m = -1e30f;
  for (int j = t; j < BATCH; j += 256) {
    float v = G[i * BATCH + j] / (ni * nrm[j]);
    G[i * BATCH + j] = v;
    m = fmaxf(m, v);
  }
  sh[t] = m; __syncthreads();
  for (int s = 128; s > 0; s >>= 1) { if (t < s) sh[t] = fmaxf(sh[t], sh[t + s]); __syncthreads(); }
  m = sh[0]; __syncthreads();
  float sum = 0.f;
  for (int j = t; j < BATCH; j += 256) {
    float e = __expf(G[i * BATCH + j] - m);
    G[i * BATCH + j] = e;
    sum += e;
  }
  sh[t] = sum; __syncthreads();
  for (int s = 128; s > 0; s >>= 1) { if (t < s) sh[t] += sh[t + s]; __syncthreads(); }
  float inv = 1.f / sh[0];
  for (int j = t; j < BATCH; j += 256) G[i * BATCH + j] *= inv;
}

// top-3 (ties -> lower index), softmax weights, weighted sum of sess rows
__global__ void k_neighbor(const float* __restrict__ P, const float* __restrict__ sess,
                           float* __restrict__ nb) {
  __shared__ float w[3];
  __shared__ int   id[3];
  int b = blockIdx.x, t = threadIdx.x;    // 128
  if (t == 0) {
    float v0 = -1e30f, v1 = -1e30f, v2 = -1e30f;
    int i0 = 0, i1 = 0, i2 = 0;
    for (int j = 0; j < BATCH; ++j) {
      float v = P[b * BATCH + j];
      if (v > v0)      { v2 = v1; i2 = i1; v1 = v0; i1 = i0; v0 = v; i0 = j; }
      else if (v > v1) { v2 = v1; i2 = i1; v1 = v;  i1 = j; }
      else if (v > v2) { v2 = v;  i2 = j; }
    }
    float e0 = 1.f, e1 = __expf(v1 - v0), e2 = __expf(v2 - v0);
    float s = e0 + e1 + e2;
    w[0] = e0 / s; w[1] = e1 / s; w[2] = e2 / s;
    id[0] = i0; id[1] = i1; id[2] = i2;
  }
  __syncthreads();
  if (t < EMB)
    nb[b * EMB + t] = w[0] * sess[id[0] * PE + t] + w[1] * sess[id[1] * PE + t]
                    + w[2] * sess[id[2] * PE + t];
}

// X += bias (cols < EMB), optional elu (alpha=1)
__global__ void k_bias_act(float* __restrict__ X, const float* __restrict__ bias,
                           int rows, int do_elu) {
  int idx = blockIdx.x * blockDim.x + threadIdx.x;
  int r = idx / PE, c = idx % PE;
  if (r >= rows || c >= EMB) return;
  float x = X[r * PE + c] + bias[c];
  if (do_elu) x = x > 0.f ? x : (__expf(x) - 1.f);
  X[r * PE + c] = x;
}

__global__ void k_rownorm(const float* __restrict__ X, float* __restrict__ n) {
  __shared__ float sh[128];
  int r = blockIdx.x, t = threadIdx.x;
  float v = (t < EMB) ? X[r * PE + t] : 0.f;
  sh[t] = v * v;
  __syncthreads();
  for (int s = 64; s > 0; s >>= 1) { if (t < s) sh[t] += sh[t + s]; __syncthreads(); }
  if (t == 0) n[r] = sqrtf(sh[0]);
}

// per-row: -log( exp(S_ii/(n1_i n2_i tau)) / (sum_j exp(S_ij/(n1_i n2_j tau)) + 1e-8) )
__global__ void k_loss_row(const float* __restrict__ S, const float* __restrict__ n1,
                           const float* __restrict__ n2, float* __restrict__ rl) {
  __shared__ float sh[256];
  int i = blockIdx.x, t = threadIdx.x;
  float a = n1[i];
  float sum = 0.f;
  for (int j = t; j < BATCH; j += 256)
    sum += __expf(S[i * BATCH + j] / (a * n2[j]) / TAU);
  sh[t] = sum; __syncthreads();
  for (int s = 128; s > 0; s >>= 1) { if (t < s) sh[t] += sh[t + s]; __syncthreads(); }
  if (t == 0) {
    float diag = __expf(S[i * BATCH + i] / (a * n2[i]) / TAU);
    rl[i] = -__logf(diag / (sh[0] + 1e-8f));
  }
}

__global__ void k_loss_reduce(const float* __restrict__ rl, float* __restrict__ out) {
  __shared__ float sh[256];
  int t = threadIdx.x;
  float s = 0.f;
  for (int j = t; j < BATCH; j += 256) s += rl[j];
  sh[t] = s; __syncthreads();
  for (int st = 128; st > 0; st >>= 1) { if (t < st) sh[t] += sh[t + st]; __syncthreads(); }
  if (t == 0) *out = sh[0] / (float)BATCH;
}

// ---------------- launch ----------------
extern "C" void kernel_launch(void* const* d_in, const int* in_sizes, int n_in,
                              void* d_out, int out_size, void* d_ws, size_t ws_size,
                              hipStream_t stream) {
  (void)in_sizes; (void)n_in; (void)out_size; (void)ws_size;
  const float* emb   = (const float*)d_in[0];
  const float* aval  = (const float*)d_in[1];
  const float* Amat  = (const float*)d_in[2];
  const float* Dmat  = (const float*)d_in[3];
  const float* z2    = (const float*)d_in[4];
  const float* Wcf1  = (const float*)d_in[5];
  const float* bcf1  = (const float*)d_in[6];
  const float* Wcf2  = (const float*)d_in[7];
  const float* bcf2  = (const float*)d_in[8];
  const float* Wkg1  = (const float*)d_in[9];
  const float* bkg1  = (const float*)d_in[10];
  const float* Wkg2  = (const float*)d_in[11];
  const float* bkg2  = (const float*)d_in[12];
  const float* slen  = (const float*)d_in[13];
  const int*   arow  = (const int*)d_in[14];
  const int*   acol  = (const int*)d_in[15];
  const int*   items = (const int*)d_in[16];

  float* out   = (float*)d_out;
  float* hyper = out;
  float* lc    = out + (size_t)N_NODE * EMB;
  float* nb    = lc + (size_t)BATCH * EMB;
  float* loss  = nb + (size_t)BATCH * EMB;

  const size_t NE = (size_t)N_NODE * EMB;
  const size_t BP = (size_t)BATCH * PE;
  float* w     = (float*)d_ws;
  float* h0    = w;                 // [N_NODE,EMB]
  float* h1    = h0 + NE;           // [N_NODE,EMB]
  float* rel   = h1 + NE;           // [B,PE]
  float* tbuf  = rel + BP;          // [B,PE]
  float* rel2  = tbuf + BP;         // [B,PE]
  float* sess  = rel2 + BP;         // [B,PE] normalized
  float* nrm   = sess + BP;         // [B]
  float* cosb  = nrm + BATCH;       // [B,B]
  float* Wc1p  = cosb + (size_t)BATCH * BATCH;   // [PE,PE]
  float* Wc2p  = Wc1p + PE * PE;
  float* Wk1p  = Wc2p + PE * PE;
  float* Wk2p  = Wk1p + PE * PE;
  float* z2pad = Wk2p + PE * PE;    // [B,PE]
  float* tmp   = z2pad + BP;        // [B,PE]
  float* z1p   = tmp + BP;          // [B,PE]
  float* z2p   = z1p + BP;          // [B,PE]
  float* n1    = z2p + BP;          // [B]
  float* n2    = n1 + BATCH;        // [B]
  float* Sb    = n2 + BATCH;        // [B,B]
  float* rl    = Sb + (size_t)BATCH * BATCH;     // [B]

  // ---- HyperConv (2 layers + average) ----
  k_memset<<<2048, 256, 0, stream>>>(h0, NE);
  k_spmm<<<NNZ_, 128, 0, stream>>>(aval, arow, acol, emb, h0);
  k_memset<<<2048, 256, 0, stream>>>(h1, NE);
  k_spmm<<<NNZ_, 128, 0, stream>>>(aval, arow, acol, h0, h1);
  k_combine<<<2048, 256, 0, stream>>>(emb, h0, h1, hyper, NE);

  // ---- RelationGAT: rel2 = D @ (A @ rel) ----
  k_gather_mean<<<BATCH, 128, 0, stream>>>(hyper, items, slen, rel);
  dim3 gSkinny(BATCH / 16, PE / 16);
  k_gemm_nn<<<gSkinny, 32, 0, stream>>>(Amat, rel, tbuf, BATCH, PE, BATCH, BATCH, PE, PE);
  k_gemm_nn<<<gSkinny, 32, 0, stream>>>(Dmat, tbuf, rel2, BATCH, PE, BATCH, BATCH, PE, PE);
  k_selu_norm<<<BATCH, 128, 0, stream>>>(rel2, lc, sess, nrm);

  // ---- FindNeighbors ----
  dim3 gSquare(BATCH / 16, BATCH / 16);
  k_gemm_nt<<<gSquare, 32, 0, stream>>>(sess, sess, cosb, BATCH, BATCH, PE, PE, PE, BATCH);
  k_cos_softmax<<<BATCH, 256, 0, stream>>>(cosb, nrm);
  k_neighbor<<<BATCH, 128, 0, stream>>>(cosb, sess, nb);

  // ---- Contrast_2view ----
  int pcw = (PE * PE + 255) / 256;
  k_pad_copy<<<pcw, 256, 0, stream>>>(Wcf1, Wc1p, EMB, EMB, PE, PE);
  k_pad_copy<<<pcw, 256, 0, stream>>>(Wcf2, Wc2p, EMB, EMB, PE, PE);
  k_pad_copy<<<pcw, 256, 0, stream>>>(Wkg1, Wk1p, EMB, EMB, PE, PE);
  k_pad_copy<<<pcw, 256, 0, stream>>>(Wkg2, Wk2p, EMB, EMB, PE, PE);
  int pcz = (int)((BP + 255) / 256);
  k_pad_copy<<<pcz, 256, 0, stream>>>(z2, z2pad, BATCH, EMB, BATCH, PE);

  // z1p = elu(sess @ Wcf1^T + b1) @ Wcf2^T + b2
  k_gemm_nt<<<gSkinny, 32, 0, stream>>>(sess, Wc1p, tmp, BATCH, PE, PE, PE, PE, PE);
  k_bias_act<<<pcz, 256, 0, stream>>>(tmp, bcf1, BATCH, 1);
  k_gemm_nt<<<gSkinny, 32, 0, stream>>>(tmp, Wc2p, z1p, BATCH, PE, PE, PE, PE, PE);
  k_bias_act<<<pcz, 256, 0, stream>>>(z1p, bcf2, BATCH, 0);
  // z2p = elu(z2 @ Wkg1^T + b1) @ Wkg2^T + b2
  k_gemm_nt<<<gSkinny, 32, 0, stream>>>(z2pad, Wk1p, tmp, BATCH, PE, PE, PE, PE, PE);
  k_bias_act<<<pcz, 256, 0, stream>>>(tmp, bkg1, BATCH, 1);
  k_gemm_nt<<<gSkinny, 32, 0, stream>>>(tmp, Wk2p, z2p, BATCH, PE, PE, PE, PE, PE);
  k_bias_act<<<pcz, 256, 0, stream>>>(z2p, bkg2, BATCH, 0);

  k_rownorm<<<BATCH, 128, 0, stream>>>(z1p, n1);
  k_rownorm<<<BATCH, 128, 0, stream>>>(z2p, n2);
  k_gemm_nt<<<gSquare, 32, 0, stream>>>(z1p, z2p, Sb, BATCH, BATCH, PE, PE, PE, BATCH);
  k_loss_row<<<BATCH, 256, 0, stream>>>(Sb, n1, n2, rl);
  k_loss_reduce<<<1, 256, 0, stream>>>(rl, loss);
}